// PoseNet_55671366091548
// MI455X (gfx1250) — compile-verified
//
#include <hip/hip_runtime.h>

// Problem constants from the reference
#define BSZ   8
#define CSZ   3
#define NSZ   4096
#define KSZ   16
#define QT    16            // queries per block (one WMMA M-tile)
#define CHUNK 512           // candidate columns buffered in LDS per pass
#define NCHUNKS (NSZ / CHUNK)
#define WAVES 16
#define THREADS (WAVES * 32)

typedef __attribute__((ext_vector_type(2))) float v2f;
typedef __attribute__((ext_vector_type(8))) float v8f;

__global__ __launch_bounds__(THREADS)
void knn_edge_kernel(const float* __restrict__ cloud, float* __restrict__ out) {
    const int blk   = blockIdx.x;
    const int b     = blk / (NSZ / QT);
    const int qtile = blk % (NSZ / QT);
    const int qbase = qtile * QT;

    const int lane = threadIdx.x & 31;
    const int wave = threadIdx.x >> 5;
    const int hi   = lane >> 4;      // half-wave select (K=0,1 vs K=2,3 operand rows)
    const int l15  = lane & 15;

    const float* __restrict__ cx = cloud + (size_t)b * CSZ * NSZ;
    const float* __restrict__ cy = cx + NSZ;
    const float* __restrict__ cz = cy + NSZ;

    __shared__ float dist_s[QT * CHUNK];   // 32 KB distance strip

    // ---- A operand: 16x4 query tile, row m = (-2x, -2y, -2z, 1) ----
    // ISA layout: lanes 0-15 hold K=0 (v0) and K=1 (v1); lanes 16-31 hold K=2,K=3.
    v2f amat;
    {
        const int qn = qbase + l15;
        const float qx = cx[qn], qy = cy[qn], qz = cz[qn];
        amat.x = hi ? (-2.0f * qz) : (-2.0f * qx);
        amat.y = hi ? 1.0f         : (-2.0f * qy);
    }

    // ---- per-lane sorted top-16 (ascending distance, stable) ----
    float dl[KSZ];
    int   il[KSZ];
#pragma unroll
    for (int j = 0; j < KSZ; ++j) { dl[j] = __builtin_inff(); il[j] = 0x7fffffff; }

    for (int ch = 0; ch < NCHUNKS; ++ch) {
        const int colbase = ch * CHUNK;

        // ---- matrix pipe: each wave fills 2 WMMA tiles of the strip ----
#pragma unroll
        for (int t = 0; t < 2; ++t) {
            const int tcol = (wave * 2 + t) * 16;
            const int cn   = colbase + tcol + l15;
            const float bx = cx[cn], by = cy[cn], bz = cz[cn];
            const float bsq = fmaf(bx, bx, fmaf(by, by, bz * bz));
            // B operand 4x16: col n = (x', y', z', sq'); half-wave K split as for A.
            v2f bmat;
            bmat.x = hi ? bz  : bx;
            bmat.y = hi ? bsq : by;
            v8f acc = {};
            acc = __builtin_amdgcn_wmma_f32_16x16x4_f32(
                false, amat, false, bmat, (short)0, acc, false, false);
            // D layout: lanes 0-15 -> rows v, N=l15 ; lanes 16-31 -> rows v+8.
#pragma unroll
            for (int v = 0; v < 8; ++v)
                dist_s[(v + hi * 8) * CHUNK + tcol + l15] = acc[v];
        }
        __syncthreads();

        // ---- selection: wave w scans row w of the strip ----
#pragma unroll
        for (int i = 0; i < CHUNK / 32; ++i) {
            const int c    = i * 32 + lane;
            const float nd = dist_s[wave * CHUNK + c];
            const int nidx = colbase + c;
            if (nd < dl[KSZ - 1]) {               // strict < keeps lower index on ties
                dl[KSZ - 1] = nd; il[KSZ - 1] = nidx;
#pragma unroll
                for (int j = KSZ - 1; j > 0; --j) {
                    if (dl[j] < dl[j - 1]) {
                        float td = dl[j]; dl[j] = dl[j - 1]; dl[j - 1] = td;
                        int   ti = il[j]; il[j] = il[j - 1]; il[j - 1] = ti;
                    }
                }
            }
        }
        __syncthreads();
    }

    // ---- cross-lane merge: 16 rounds of wave-argmin (tie -> lowest index) ----
    int p = 0;
    int sel = -1;
#pragma unroll
    for (int k = 0; k < KSZ; ++k) {
        float v  = dl[p];
        int   ii = il[p];
        const int myhead = ii;
#pragma unroll
        for (int s = 16; s >= 1; s >>= 1) {
            const float ov = __shfl_xor(v, s, 32);
            const int   oi = __shfl_xor(ii, s, 32);
            if (ov < v || (ov == v && oi < ii)) { v = ov; ii = oi; }
        }
        if (myhead == ii) ++p;       // winner lane advances its sorted list
        if (lane == k) sel = ii;     // lane k keeps k-th nearest index
    }

    // ---- emit edge features: out[b, 0:3, n, k] = central ; out[b, 3:6, n, k] = nbr - central
    const int n = qbase + wave;
    if (lane < KSZ) {
        const float qx0 = cx[n], qy0 = cy[n], qz0 = cz[n];
        const int m = sel;
        const float nx = cx[m], ny = cy[m], nz = cz[m];
        const size_t cs   = (size_t)NSZ * KSZ;
        const size_t base = (size_t)b * 6 * cs + (size_t)n * KSZ + (size_t)lane;
        out[base + 0 * cs] = qx0;
        out[base + 1 * cs] = qy0;
        out[base + 2 * cs] = qz0;
        out[base + 3 * cs] = nx - qx0;
        out[base + 4 * cs] = ny - qy0;
        out[base + 5 * cs] = nz - qz0;
    }
}

extern "C" void kernel_launch(void* const* d_in, const int* in_sizes, int n_in,
                              void* d_out, int out_size, void* d_ws, size_t ws_size,
                              hipStream_t stream) {
    (void)in_sizes; (void)n_in; (void)out_size; (void)d_ws; (void)ws_size;
    const float* cloud = (const float*)d_in[0];
    float* out = (float*)d_out;
    const dim3 grid(BSZ * (NSZ / QT));   // 2048 workgroups
    knn_edge_kernel<<<grid, THREADS, 0, stream>>>(cloud, out);
}